// SubjCRFModel_84696755077782
// MI455X (gfx1250) — compile-verified
//
#include <hip/hip_runtime.h>
#include <stdint.h>

// Problem constants (from reference): B=128, S=1024, D=512, LABELS=32, T=34
#define Bn 128
#define Sn 1024
#define Dn 512
#define Tn 34
#define START_TAG 32
#define STOP_TAG 33

typedef float v2f __attribute__((ext_vector_type(2)));
typedef float v8f __attribute__((ext_vector_type(8)));

// ---------------------------------------------------------------------------
// Kernel 1: emissions = hidden @ W^T + b   via V_WMMA_F32_16X16X4_F32
//   hidden: [B*S, D] fp32 (streamed ONCE, ~256 MB => HBM-bound; loaded with
//           non-temporal hint so the stream does not sweep the 192 MB L2 and
//           evict emis/bp, which the later kernels re-read)
//   W:      [T, D]  fp32, staged in LDS as pre-swizzled B fragments
//   emis:   [B*S, T] fp32 (default RT policy -> stays L2-resident, 17.8 MB)
// Block = 256 threads (8 waves). Each wave owns a 16-row block; block covers
// 128 rows. 3 N-tiles (48 cols, cols >= 34 masked on store). K loop = 128.
// ---------------------------------------------------------------------------
__launch_bounds__(256)
__global__ void emis_gemm(const float* __restrict__ hidden,
                          const float* __restrict__ W,
                          const float* __restrict__ bias,
                          float* __restrict__ emis) {
  // LDS: B fragments, [kk=128][nt=3][lane=32] of v2f = 96 KB
  __shared__ v2f ldsW[128 * 3 * 32];

  const int tid = threadIdx.x;

  // Stage W into LDS in WMMA B-fragment layout:
  //   lane<16 holds {K0,K1}, lane>=16 holds {K2,K3} of the 4-row B tile,
  //   N = nt*16 + (lane&15). B columns come from rows of W (W is [T][D]).
  for (int idx = tid; idx < 128 * 96; idx += 256) {
    const int kk   = idx / 96;
    const int rem  = idx - kk * 96;
    const int nt   = rem >> 5;
    const int lane = rem & 31;
    const int n    = nt * 16 + (lane & 15);
    const int k    = 4 * kk + 2 * (lane >> 4);
    v2f v; v.x = 0.0f; v.y = 0.0f;
    if (n < Tn) { v.x = W[n * Dn + k]; v.y = W[n * Dn + k + 1]; }
    ldsW[idx] = v;
  }
  __syncthreads();

  const int wave = tid >> 5;
  const int lane = tid & 31;
  const int hi   = lane >> 4;      // selects K pair {0,1} vs {2,3}
  const int cl   = lane & 15;

  const size_t rowblk = (size_t)blockIdx.x * 128 + (size_t)wave * 16;
  // A fragment (16x4 fp32): M = lane&15, v0 = K(2*hi), v1 = K(2*hi+1)
  const float* aptr = hidden + (rowblk + (size_t)cl) * Dn + 2 * hi;

  v8f acc0 = {}, acc1 = {}, acc2 = {};
  for (int kk = 0; kk < 128; ++kk) {
    // Non-temporal: hidden is a one-shot 256 MB stream; keep it out of L2's
    // retained set so emis/bp stay resident for the following kernels.
    v2f a = __builtin_nontemporal_load((const v2f*)(aptr + kk * 4));
    const v2f* bw = &ldsW[kk * 96 + lane];
    v2f b0 = bw[0];
    v2f b1 = bw[32];
    v2f b2 = bw[64];
    acc0 = __builtin_amdgcn_wmma_f32_16x16x4_f32(false, a, false, b0,
                                                 (short)0, acc0, false, false);
    acc1 = __builtin_amdgcn_wmma_f32_16x16x4_f32(false, a, false, b1,
                                                 (short)0, acc1, false, false);
    acc2 = __builtin_amdgcn_wmma_f32_16x16x4_f32(false, a, false, b2,
                                                 (short)0, acc2, false, false);
  }

  // C/D layout: VGPR i -> M = i + 8*(lane>>4), N = lane&15 (+ nt*16)
  #pragma unroll
  for (int i = 0; i < 8; ++i) {
    const int m = i + 8 * hi;
    float* orow = emis + (rowblk + (size_t)m) * Tn;
    orow[cl]      = acc0[i] + bias[cl];
    orow[16 + cl] = acc1[i] + bias[16 + cl];
    if (32 + cl < Tn) orow[32 + cl] = acc2[i] + bias[32 + cl];
  }
}

// ---------------------------------------------------------------------------
// Kernel 2: Viterbi forward. One block per batch, 64 threads (tag j = tid).
// part[] double-buffered in LDS; transitions in LDS; bp stored as uint8.
// Captures part at t == len-1 (last_part) and final STOP argmax (pointer).
// Strict '>' keeps the FIRST max index, matching jnp.argmax.
// ---------------------------------------------------------------------------
__launch_bounds__(64)
__global__ void viterbi_forward(const float* __restrict__ emis,
                                const unsigned char* __restrict__ mask,
                                const float* __restrict__ trans,
                                unsigned char* __restrict__ bp,
                                int* __restrict__ ptr_out,
                                int* __restrict__ len_out) {
  __shared__ float ltr[Tn * Tn];
  __shared__ float part[2][Tn];
  __shared__ float lp[Tn];
  __shared__ int   sred[64];

  const int j = threadIdx.x;
  const int b = blockIdx.x;

  for (int idx = j; idx < Tn * Tn; idx += 64) ltr[idx] = trans[idx];

  // length[b] = sum(mask[b, :]) ; mask is 1 byte per element (jax bool)
  int acc = 0;
  const unsigned char* mrow = mask + (size_t)b * Sn + (size_t)j * 16;
  #pragma unroll
  for (int i = 0; i < 16; ++i) acc += mrow[i];
  sred[j] = acc;
  __syncthreads();
  for (int off = 32; off > 0; off >>= 1) {
    if (j < off) sred[j] += sred[j + off];
    __syncthreads();
  }
  const int len = sred[0];

  const float* ebase = emis + (size_t)b * Sn * Tn;

  if (j < Tn) {
    const float p0 = ltr[START_TAG * Tn + j] + ebase[j];
    part[0][j] = p0;
    if (len == 1) lp[j] = p0;
  }
  __syncthreads();

  for (int t = 1; t < Sn; ++t) {
    const int src = (t - 1) & 1;
    const int dst = t & 1;
    if (j < Tn) {
      float best = -3.4e38f;
      int   bi   = 0;
      #pragma unroll
      for (int p = 0; p < Tn; ++p) {
        const float v = part[src][p] + ltr[p * Tn + j];
        if (v > best) { best = v; bi = p; }
      }
      const float np = best + ebase[(size_t)t * Tn + j];
      part[dst][j] = np;
      // back_points row s = t-1, zeroed when t >= len (reference mask)
      bp[((size_t)b * Sn + (size_t)(t - 1)) * Tn + j] =
          (t < len) ? (unsigned char)bi : (unsigned char)0;
      if (t == len - 1) lp[j] = np;
    }
    __syncthreads();
  }

  if (j == 0) {
    float best = -3.4e38f;
    int   bi   = 0;
    for (int p = 0; p < Tn; ++p) {
      const float v = lp[p] + ltr[p * Tn + STOP_TAG];
      if (v > best) { best = v; bi = p; }
    }
    ptr_out[b] = bi;
    len_out[b] = len;
  }
}

// ---------------------------------------------------------------------------
// Kernel 3: backtrace. One block per batch. Stage the batch's 34 KB bp slab
// into LDS with uint4 loads, then a single lane chases pointers through LDS
// (~25 cyc/hop vs ~500 cyc/hop through HBM).
//   out[b][S-1] = pointer;  for s=S-2..0: p = (s==len-1)? pointer : bp[s][p]
// This reproduces the reference's back_points[len-1] = pointer overwrite and
// the zero rows past the sequence length.
// ---------------------------------------------------------------------------
__launch_bounds__(256)
__global__ void viterbi_backtrace(const unsigned char* __restrict__ bp,
                                  const int* __restrict__ ptr_in,
                                  const int* __restrict__ len_in,
                                  int* __restrict__ out) {
  __shared__ uint4 sbp4[(Sn * Tn) / 16];   // 34816 bytes

  const int tid = threadIdx.x;
  const int b   = blockIdx.x;

  const uint4* src = (const uint4*)(bp + (size_t)b * Sn * Tn);
  for (int idx = tid; idx < (Sn * Tn) / 16; idx += 256) sbp4[idx] = src[idx];
  __syncthreads();

  if (tid == 0) {
    const unsigned char* sbp = (const unsigned char*)sbp4;
    const int p0  = ptr_in[b];
    const int len = len_in[b];
    int* orow = out + (size_t)b * Sn;
    int p = p0;
    orow[Sn - 1] = p;
    for (int s = Sn - 2; s >= 0; --s) {
      p = (s == len - 1) ? p0 : (int)sbp[s * Tn + p];
      orow[s] = p;
    }
  }
}

// ---------------------------------------------------------------------------
extern "C" void kernel_launch(void* const* d_in, const int* in_sizes, int n_in,
                              void* d_out, int out_size, void* d_ws,
                              size_t ws_size, hipStream_t stream) {
  (void)in_sizes; (void)n_in; (void)out_size; (void)ws_size;

  const float*         hidden = (const float*)d_in[0];          // [B,S,D]
  const unsigned char* mask   = (const unsigned char*)d_in[1];  // [B,S] bool
  const float*         W      = (const float*)d_in[2];          // [T,D]
  const float*         bias   = (const float*)d_in[3];          // [T]
  const float*         trans  = (const float*)d_in[4];          // [T,T]
  int*                 out    = (int*)d_out;                    // [B,S] int32

  char* ws = (char*)d_ws;
  const size_t emis_bytes = (size_t)Bn * Sn * Tn * sizeof(float); // 17.8 MB
  const size_t bp_bytes   = (size_t)Bn * Sn * Tn;                 //  4.4 MB
  float*         emis   = (float*)ws;
  unsigned char* bpbuf  = (unsigned char*)(ws + emis_bytes);
  int*           ptrbuf = (int*)(ws + emis_bytes + bp_bytes);
  int*           lenbuf = ptrbuf + Bn;

  // 1024 blocks x 256 threads; each block covers 128 rows of B*S = 131072
  emis_gemm<<<(Bn * Sn) / 128, 256, 0, stream>>>(hidden, W, bias, emis);
  viterbi_forward<<<Bn, 64, 0, stream>>>(emis, mask, trans, bpbuf, ptrbuf,
                                         lenbuf);
  viterbi_backtrace<<<Bn, 256, 0, stream>>>(bpbuf, ptrbuf, lenbuf, out);
}